// GIN_MLP_Encoder_18056042513111
// MI455X (gfx1250) — compile-verified
//
#include <hip/hip_runtime.h>

#define NN 100000
#define EE 1600000
#define FIN 128
#define DIM 64

typedef __attribute__((ext_vector_type(2))) float v2f;
typedef __attribute__((ext_vector_type(8))) float v8f;

// ---------------------------------------------------------------------------
// Native f32 global atomic add (GLOBAL_ATOMIC_ADD_F32, no-return form).
// ---------------------------------------------------------------------------
__device__ __forceinline__ void atomic_add_f32(float* p, float v) {
    asm volatile("global_atomic_add_f32 %0, %1, off" : : "v"(p), "v"(v) : "memory");
}

// ---------------------------------------------------------------------------
// zero a buffer (float4 granularity)
// ---------------------------------------------------------------------------
__global__ void zero4(float* __restrict__ p, int n4) {
    int t = blockIdx.x * blockDim.x + threadIdx.x;
    if (t < n4) ((float4*)p)[t] = make_float4(0.f, 0.f, 0.f, 0.f);
}

// ---------------------------------------------------------------------------
// agg[dst[e]] += x[src[e]]  (segment_sum).  One thread per (edge, 4-feature
// chunk).  ei layout: [0:E)=src, [E:2E)=dst.
// ---------------------------------------------------------------------------
__global__ void scatter_add(const float* __restrict__ x, const int* __restrict__ ei,
                            float* __restrict__ agg, int E, int F) {
    int t = blockIdx.x * blockDim.x + threadIdx.x;
    int chunks = F >> 2;
    if (t >= E * chunks) return;
    int e = t / chunks;
    int c = (t - e * chunks) << 2;
    int s = ei[e];
    int d = ei[E + e];
    float4 v = *(const float4*)(x + (size_t)s * F + c);
    float* p = agg + (size_t)d * F + c;
    atomic_add_f32(p + 0, v.x);
    atomic_add_f32(p + 1, v.y);
    atomic_add_f32(p + 2, v.z);
    atomic_add_f32(p + 3, v.w);
}

// ---------------------------------------------------------------------------
// WMMA GEMM: out[N x 64] = act( (A (+AGG)) [N x K] * W [K x 64] + bias )
// One wave per 16-row x 64-col strip; 4 accumulator tiles; f32 WMMA 16x16x4.
// Weights staged in LDS.
//
// f32 A-frag (16x4), lane L: row = L&15, holds K = k0+2*(L>>4) and +1.
// f32 B-frag (4x16), lane L: col = L&15, VGPR0 holds K = k0+2*(L>>4), VGPR1 +1.
// f32 D (16x16), VGPR r, lane L: row = r + 8*(L>>4), col = L&15.
// ---------------------------------------------------------------------------
template<int K, bool ADD_AGG, bool RELU>
__global__ __launch_bounds__(256) void gemm16(const float* __restrict__ A,
                                              const float* __restrict__ AGG,
                                              const float* __restrict__ W,
                                              const float* __restrict__ bias,
                                              float* __restrict__ out, int nrows) {
    __shared__ float sW[K * DIM];
    for (int i = threadIdx.x; i < K * DIM; i += 256) sW[i] = W[i];
    __syncthreads();

    const int lane = threadIdx.x & 31;
    const int wave = threadIdx.x >> 5;
    const int tileRow = (blockIdx.x * 8 + wave) * 16;
    if (tileRow >= nrows) return;

    const int m  = lane & 15;        // row (A) / col (B) within tile
    const int kk = (lane >> 4) * 2;  // k sub-offset: 0 or 2

    v8f acc[4];
    acc[0] = v8f{}; acc[1] = v8f{}; acc[2] = v8f{}; acc[3] = v8f{};

    const float* Arow = A + (size_t)(tileRow + m) * K;
    const float* Grow = ADD_AGG ? (AGG + (size_t)(tileRow + m) * K) : nullptr;

    for (int k0 = 0; k0 < K; k0 += 4) {
        float2 av = *(const float2*)(Arow + k0 + kk);
        if (ADD_AGG) {
            float2 gv = *(const float2*)(Grow + k0 + kk);
            av.x += gv.x; av.y += gv.y;
        }
        v2f a; a.x = av.x; a.y = av.y;
        const float* w0 = &sW[(k0 + kk) * DIM];
        const float* w1 = &sW[(k0 + kk + 1) * DIM];
#pragma unroll
        for (int ct = 0; ct < 4; ++ct) {
            v2f b;
            b.x = w0[ct * 16 + m];
            b.y = w1[ct * 16 + m];
            acc[ct] = __builtin_amdgcn_wmma_f32_16x16x4_f32(
                false, a, false, b, (short)0, acc[ct], false, false);
        }
    }

    const int rowHalf = (lane >> 4) * 8;
#pragma unroll
    for (int ct = 0; ct < 4; ++ct) {
        const float bs = bias[ct * 16 + m];
#pragma unroll
        for (int r = 0; r < 8; ++r) {
            int row = tileRow + r + rowHalf;
            float v = acc[ct][r] + bs;
            if (RELU) v = v > 0.f ? v : 0.f;
            out[(size_t)row * DIM + ct * 16 + m] = v;
        }
    }
}

// ---------------------------------------------------------------------------
// BatchNorm stats: per-channel sum and sum-of-squares (64 channels).
// blockDim = 64 (one channel per thread), grid = row partitions.
// ---------------------------------------------------------------------------
__global__ void bn_stats(const float* __restrict__ h, float* __restrict__ stats, int nrows) {
    int c = threadIdx.x;
    float s = 0.f, s2 = 0.f;
    for (int r = blockIdx.x; r < nrows; r += gridDim.x) {
        float v = h[(size_t)r * DIM + c];
        s += v; s2 += v * v;
    }
    atomic_add_f32(&stats[c], s);
    atomic_add_f32(&stats[DIM + c], s2);
}

__global__ void bn_norm(float* __restrict__ h, const float* __restrict__ stats,
                        const float* __restrict__ g, const float* __restrict__ b, int nrows) {
    int t = blockIdx.x * blockDim.x + threadIdx.x;
    if (t >= nrows * DIM) return;
    int c = t & (DIM - 1);
    float inv_n = 1.f / (float)nrows;
    float mu  = stats[c] * inv_n;
    float var = stats[DIM + c] * inv_n - mu * mu;
    h[t] = g[c] * (h[t] - mu) * rsqrtf(var + 1e-5f) + b[c];
}

// ---------------------------------------------------------------------------
// Edge head: out[e] = concat(h[src], h[dst]) @ fc_w + fc_b   ([E x 2])
// fc_w layout: [128][2] row-major.
// ---------------------------------------------------------------------------
__global__ void edge_head(const float* __restrict__ h, const int* __restrict__ ei,
                          const float* __restrict__ fw, const float* __restrict__ fb,
                          float* __restrict__ out, int E) {
    int e = blockIdx.x * blockDim.x + threadIdx.x;
    if (e >= E) return;
    int s = ei[e];
    int d = ei[E + e];
    const float* hs = h + (size_t)s * DIM;
    const float* hd = h + (size_t)d * DIM;
    float a0 = fb[0], a1 = fb[1];
#pragma unroll 8
    for (int i = 0; i < DIM; ++i) {
        float v = hs[i];
        a0 += v * fw[2 * i];
        a1 += v * fw[2 * i + 1];
    }
#pragma unroll 8
    for (int i = 0; i < DIM; ++i) {
        float v = hd[i];
        a0 += v * fw[2 * (DIM + i)];
        a1 += v * fw[2 * (DIM + i) + 1];
    }
    out[2 * e]     = a0;
    out[2 * e + 1] = a1;
}

// ---------------------------------------------------------------------------
extern "C" void kernel_launch(void* const* d_in, const int* in_sizes, int n_in,
                              void* d_out, int out_size, void* d_ws, size_t ws_size,
                              hipStream_t stream) {
    (void)in_sizes; (void)n_in; (void)out_size; (void)ws_size;

    const float* x  = (const float*)d_in[0];
    const int*   ei = (const int*)d_in[1];
    const float* fcw = (const float*)d_in[32];
    const float* fcb = (const float*)d_in[33];

    float* ws    = (float*)d_ws;
    float* agg   = ws;                                  // N*128 f32
    float* mid   = agg + (size_t)NN * FIN;              // N*64 f32
    float* hbuf  = mid + (size_t)NN * DIM;              // N*64 f32
    float* stats = hbuf + (size_t)NN * DIM;             // 128 f32

    const int gemmBlocks = (NN / 16 + 7) / 8;           // 8 waves/block, 16 rows/wave
    const float* xin = x;

    for (int L = 0; L < 5; ++L) {
        const int Fin = (L == 0) ? FIN : DIM;
        const float* w1 = (const float*)d_in[2 + L * 6 + 0];
        const float* b1 = (const float*)d_in[2 + L * 6 + 1];
        const float* w2 = (const float*)d_in[2 + L * 6 + 2];
        const float* b2 = (const float*)d_in[2 + L * 6 + 3];
        const float* g  = (const float*)d_in[2 + L * 6 + 4];
        const float* be = (const float*)d_in[2 + L * 6 + 5];

        // agg = 0; agg[dst] += x[src]
        int n4 = (NN * Fin) >> 2;
        zero4<<<(n4 + 255) / 256, 256, 0, stream>>>(agg, n4);
        int sthreads = EE * (Fin >> 2);
        scatter_add<<<(sthreads + 255) / 256, 256, 0, stream>>>(xin, ei, agg, EE, Fin);

        // mid = ReLU((x+agg) @ w1 + b1) ; h = ReLU(mid @ w2 + b2)
        if (Fin == FIN)
            gemm16<FIN, true, true><<<gemmBlocks, 256, 0, stream>>>(xin, agg, w1, b1, mid, NN);
        else
            gemm16<DIM, true, true><<<gemmBlocks, 256, 0, stream>>>(xin, agg, w1, b1, mid, NN);
        gemm16<DIM, false, true><<<gemmBlocks, 256, 0, stream>>>(mid, nullptr, w2, b2, hbuf, NN);

        // BatchNorm (training stats, biased variance), in place
        zero4<<<1, 32, 0, stream>>>(stats, 32);
        bn_stats<<<512, 64, 0, stream>>>(hbuf, stats, NN);
        bn_norm<<<(NN * DIM + 255) / 256, 256, 0, stream>>>(hbuf, stats, g, be, NN);

        xin = hbuf;
    }

    edge_head<<<(EE + 255) / 256, 256, 0, stream>>>(hbuf, ei, fcw, fcb, (float*)d_out, EE);
}